// SwitchingRNN_4054449127909
// MI455X (gfx1250) — compile-verified
//
#include <hip/hip_runtime.h>

#define B_  64
#define T_  512
#define I_  256
#define L_  512
#define K_  8
#define RES_J 128            // Whh rows resident in LDS (256 KB)

typedef float v2f __attribute__((ext_vector_type(2)));
typedef float v8f __attribute__((ext_vector_type(8)));

// ---------------------------------------------------------------------------
// Kernel 1a: Wih_hat[b][o][i] = sum_k p[b][k] * W_ih[k*L + o][i]
// ---------------------------------------------------------------------------
__global__ void k_mix_wih(const float* __restrict__ W_ih,
                          const float* __restrict__ p,
                          float* __restrict__ Wih_hat) {
    int e = blockIdx.x * blockDim.x + threadIdx.x;      // 64*512*256 elements
    int b = e >> 17;
    int r = e & ((1 << 17) - 1);
    int o = r >> 8;
    int i = r & 255;
    const float* pb = p + b * K_;
    float s = 0.f;
#pragma unroll
    for (int k = 0; k < K_; ++k)
        s = fmaf(pb[k], W_ih[((size_t)(k * L_ + o)) * I_ + i], s);
    Wih_hat[e] = s;
}

// ---------------------------------------------------------------------------
// Kernel 1b: WhhT[b][j][i] = sum_k p[b][k] * W_hh[k*L + i][j]
// (transposed so the scan kernel reads columns coalesced)
// ---------------------------------------------------------------------------
__global__ void k_mix_whh_T(const float* __restrict__ W_hh,
                            const float* __restrict__ p,
                            float* __restrict__ WhhT) {
    __shared__ float tile[32][33];
    int b  = blockIdx.z;
    int i0 = blockIdx.y * 32;
    int j0 = blockIdx.x * 32;
    int tx = threadIdx.x & 31;
    int ty = threadIdx.x >> 5;
    const float* pb = p + b * K_;
#pragma unroll
    for (int s = 0; s < 4; ++s) {
        int i = i0 + ty + 8 * s;
        float acc = 0.f;
#pragma unroll
        for (int k = 0; k < K_; ++k)
            acc = fmaf(pb[k], W_hh[((size_t)(k * L_ + i)) * L_ + (j0 + tx)], acc);
        tile[ty + 8 * s][tx] = acc;
    }
    __syncthreads();
#pragma unroll
    for (int s = 0; s < 4; ++s) {
        int j = j0 + ty + 8 * s;
        WhhT[((size_t)b * L_ + j) * L_ + (i0 + tx)] = tile[tx][ty + 8 * s];
    }
}

// ---------------------------------------------------------------------------
// Kernel 1c: c_hat[b][l] = sum_k p[b][k] * (b_ih + b_hh + bias)[k*L + l]
// ---------------------------------------------------------------------------
__global__ void k_mix_bias(const float* __restrict__ b_ih,
                           const float* __restrict__ b_hh,
                           const float* __restrict__ bias,
                           const float* __restrict__ p,
                           float* __restrict__ c_hat) {
    int e = blockIdx.x * blockDim.x + threadIdx.x;
    int b = e >> 9;
    int l = e & (L_ - 1);
    const float* pb = p + b * K_;
    float s = 0.f;
#pragma unroll
    for (int k = 0; k < K_; ++k) {
        int idx = k * L_ + l;
        s = fmaf(pb[k], b_ih[idx] + b_hh[idx] + bias[idx], s);
    }
    c_hat[e] = s;
}

// ---------------------------------------------------------------------------
// Kernel 2: x_hat[b][t][o] = input[b][t][:] . Wih_hat[b][o][:] + c_hat[b][o]
// via V_WMMA_F32_16X16X4_F32, written in place into d_out's states region.
// ---------------------------------------------------------------------------
__global__ void k_xproj_wmma(const float* __restrict__ input,
                             const float* __restrict__ Wih_hat,
                             const float* __restrict__ c_hat,
                             float* __restrict__ states) {
    int b    = blockIdx.y;
    int wave = threadIdx.x >> 5;
    int lane = threadIdx.x & 31;
    int tile = blockIdx.x * 8 + wave;
    int t0 = (tile >> 5) << 4;
    int o0 = (tile & 31) << 4;
    int lh = lane & 15;
    int hi = lane >> 4;

    const float* aRow = input   + ((size_t)b * T_ + (t0 + lh)) * I_;
    const float* bRow = Wih_hat + ((size_t)b * L_ + (o0 + lh)) * I_;

    float ci = c_hat[b * L_ + o0 + lh];
    v8f c = {ci, ci, ci, ci, ci, ci, ci, ci};

#pragma unroll 8
    for (int i0 = 0; i0 < I_; i0 += 4) {
        v2f a  = *(const v2f*)(aRow + i0 + 2 * hi);
        v2f bb = *(const v2f*)(bRow + i0 + 2 * hi);
        c = __builtin_amdgcn_wmma_f32_16x16x4_f32(
                false, a, false, bb, (short)0, c, false, false);
    }

    float* outp = states + ((size_t)b * T_ + t0) * L_ + o0 + lh;
#pragma unroll
    for (int v = 0; v < 8; ++v) {
        int t = v + 8 * hi;
        outp[(size_t)t * L_] = c[v];
    }
}

// ---------------------------------------------------------------------------
// Kernel 3: sequential scan, one WG (512 thr) per batch.
//  - threads = (jq in [0,4)) x (i4 in [0,128)): thread accumulates a float4
//    partial dot (output cols 4*i4..4*i4+3) over j in its quarter.
//  - quarter jq==0 of WhhT is LDS-resident (preloaded via CDNA5
//    global_load_async_to_lds_b128), quarters 1..3 stream b128 from L2.
//  - LDS float4 reduction combines quarters; h kept in LDS (broadcast).
// ---------------------------------------------------------------------------
__global__ void __launch_bounds__(512)
k_scan(const float* __restrict__ WhhT,
       const float* __restrict__ h0,
       float* __restrict__ out) {
    __shared__ float  lds_w[RES_J * L_];     // 256 KB resident weight block
    __shared__ float  hs[L_];                // 2 KB hidden state
    __shared__ float4 red[4][L_ / 4];        // 8 KB reduction buffer

    const int b   = blockIdx.x;
    const int tid = threadIdx.x;
    const int jq  = tid >> 7;                // j-quarter (wave-uniform)
    const int i4  = tid & 127;               // output quad index

    const float* Wb = WhhT + (size_t)b * L_ * L_;
    float* st = out + (size_t)b * T_ * L_;

    // async preload of rows [0, RES_J) into LDS
    {
        const float* src = Wb;
#pragma unroll
        for (int s = 0; s < (RES_J * L_ / 4) / 512; ++s) {
            int e4 = s * 512 + tid;                          // float4 index
            unsigned lds_b = (unsigned)(size_t)&lds_w[e4 * 4];
            const float* g = src + (size_t)e4 * 4;
            asm volatile("global_load_async_to_lds_b128 %0, %1, off"
                         :: "v"(lds_b), "v"(g) : "memory");
        }
        asm volatile("s_wait_asynccnt 0" ::: "memory");
    }
    hs[tid] = h0[b * L_ + tid];
    __syncthreads();

    const float4* gw  = (const float4*)(Wb + (size_t)jq * RES_J * L_) + i4;
    const float4* lw  = (const float4*)lds_w + i4;
    const float*  hq  = hs + jq * RES_J;
    float4* st4 = (float4*)st;
    float4* hs4 = (float4*)hs;

    for (int t = 0; t < T_; ++t) {
        float ax = 0.f, ay = 0.f, az = 0.f, aw = 0.f;
        if (jq == 0) {
#pragma unroll 8
            for (int j = 0; j < RES_J; ++j) {
                float4 w = lw[j * (L_ / 4)];
                float  h = hq[j];
                ax = fmaf(w.x, h, ax); ay = fmaf(w.y, h, ay);
                az = fmaf(w.z, h, az); aw = fmaf(w.w, h, aw);
            }
        } else {
#pragma unroll 8
            for (int j = 0; j < RES_J; ++j) {
                float4 w = gw[j * (L_ / 4)];
                float  h = hq[j];
                ax = fmaf(w.x, h, ax); ay = fmaf(w.y, h, ay);
                az = fmaf(w.z, h, az); aw = fmaf(w.w, h, aw);
            }
        }
        float4 a4; a4.x = ax; a4.y = ay; a4.z = az; a4.w = aw;
        red[jq][i4] = a4;
        __syncthreads();                         // partials ready; hs reads done
        if (jq == 0) {
            float4 r0 = red[0][i4], r1 = red[1][i4], r2 = red[2][i4], r3 = red[3][i4];
            float4 xq = st4[(size_t)t * (L_ / 4) + i4];   // x_hat (includes c_hat)
            float4 h4;
            h4.x = xq.x + r0.x + r1.x + r2.x + r3.x;
            h4.y = xq.y + r0.y + r1.y + r2.y + r3.y;
            h4.z = xq.z + r0.z + r1.z + r2.z + r3.z;
            h4.w = xq.w + r0.w + r1.w + r2.w + r3.w;
            hs4[i4] = h4;
            st4[(size_t)t * (L_ / 4) + i4] = h4;          // states[b][t] = h_t
        }
        __syncthreads();                         // new hs visible for t+1
    }
    out[(size_t)B_ * T_ * L_ + (size_t)b * L_ + tid] = hs[tid];   // hidden
}

// ---------------------------------------------------------------------------
extern "C" void kernel_launch(void* const* d_in, const int* in_sizes, int n_in,
                              void* d_out, int out_size, void* d_ws, size_t ws_size,
                              hipStream_t stream) {
    const float* input = (const float*)d_in[0];
    const float* h0    = (const float*)d_in[1];
    const float* p     = (const float*)d_in[2];
    const float* W_ih  = (const float*)d_in[3];
    const float* b_ih  = (const float*)d_in[4];
    const float* W_hh  = (const float*)d_in[5];
    const float* b_hh  = (const float*)d_in[6];
    const float* bias  = (const float*)d_in[7];
    float* out = (float*)d_out;

    float* WhhT    = (float*)d_ws;                       // 64 MB
    float* Wih_hat = WhhT + (size_t)B_ * L_ * L_;        // 32 MB
    float* c_hat   = Wih_hat + (size_t)B_ * L_ * I_;     // 128 KB

    k_mix_wih   <<<(B_ * L_ * I_) / 256, 256, 0, stream>>>(W_ih, p, Wih_hat);
    k_mix_whh_T <<<dim3(L_ / 32, L_ / 32, B_), 256, 0, stream>>>(W_hh, p, WhhT);
    k_mix_bias  <<<(B_ * L_) / 256, 256, 0, stream>>>(b_ih, b_hh, bias, p, c_hat);
    k_xproj_wmma<<<dim3(128, B_), 256, 0, stream>>>(input, Wih_hat, c_hat, out);
    k_scan      <<<B_, 512, 0, stream>>>(WhhT, h0, out);
}